// LightweightKGEncoder_51153060495543
// MI455X (gfx1250) — compile-verified
//
#include <hip/hip_runtime.h>
#include <hip/hip_bf16.h>

#define N_NODES 50000
#define N_EDGES 1600000
#define NODE_DIM 1024
#define HID 128
#define LN_EPS 1e-5f

typedef __attribute__((ext_vector_type(2))) float v2f;
typedef __attribute__((ext_vector_type(4))) float v4f;
typedef __attribute__((ext_vector_type(8))) float v8f;

__device__ __forceinline__ v8f wmma_f32(v2f a, v2f b, v8f c) {
    return __builtin_amdgcn_wmma_f32_16x16x4_f32(
        /*neg_a=*/false, a, /*neg_b=*/false, b,
        /*c_mod=*/(short)0, c, /*reuse_a=*/false, /*reuse_b=*/false);
}

__device__ __forceinline__ v2f lo2(v4f v) { return __builtin_shufflevector(v, v, 0, 1); }
__device__ __forceinline__ v2f hi2(v4f v) { return __builtin_shufflevector(v, v, 2, 3); }

__device__ __forceinline__ void gatomic_add(float* p, float v) {
    __hip_atomic_fetch_add(p, v, __ATOMIC_RELAXED, __HIP_MEMORY_SCOPE_AGENT);
}

// ---------------------------------------------------------------------------
// Kernel 1: h = relu(x @ proj_w^T + proj_b)   [N, 1024] -> [N, 128]
// One block = 16 rows; wave w computes the 16x16 tile at columns j0 = w*16.
// K ordering is permuted so each half-wave owns a 4-wide K chunk inside an
// 8-wide step: one global_load_b128 per operand feeds TWO wmma ops.
// ---------------------------------------------------------------------------
__global__ __launch_bounds__(256) void k_proj(const float* __restrict__ x,
                                              const float* __restrict__ w,
                                              const float* __restrict__ bias,
                                              float* __restrict__ h) {
    const int lane  = threadIdx.x & 31;
    const int wave  = threadIdx.x >> 5;
    const int row0  = blockIdx.x * 16;
    const int j0    = wave * 16;
    const int m     = lane & 15;           // A row / B col within tile
    const int koff4 = (lane >> 4) * 4;     // half-wave K base inside 8-wide chunk

    const float* xr = x + (size_t)(row0 + m) * NODE_DIM + koff4;
    const float* wr = w + (size_t)(j0 + m) * NODE_DIM + koff4;

    v8f c = {0.f, 0.f, 0.f, 0.f, 0.f, 0.f, 0.f, 0.f};
#pragma unroll 4
    for (int k = 0; k < NODE_DIM; k += 8) {
        v4f a4 = *(const v4f*)(xr + k);    // K = k+koff4 .. +3
        v4f b4 = *(const v4f*)(wr + k);
        c = wmma_f32(lo2(a4), lo2(b4), c); // slots: K = k+{0,1, 4,5}
        c = wmma_f32(hi2(a4), hi2(b4), c); // slots: K = k+{2,3, 6,7}
    }

    const int   col = j0 + m;
    const float bv  = bias[col];
    const int   rb  = row0 + (lane >> 4) * 8;
#pragma unroll
    for (int v = 0; v < 8; ++v) {
        float val = c[v] + bv;
        h[(size_t)(rb + v) * HID + col] = val > 0.f ? val : 0.f;
    }
}

// ---------------------------------------------------------------------------
// Kernel 2: edge scatter: agg[dst] += h[src]; deg[dst] += 1
// One wave (32 lanes) per edge; each lane handles 4 features (float4 gather).
// h and agg are L2-resident (25.6 MB each vs 192 MB L2).
// ---------------------------------------------------------------------------
__global__ __launch_bounds__(256) void k_scatter(const int* __restrict__ ei,
                                                 const float* __restrict__ h,
                                                 float* __restrict__ agg,
                                                 float* __restrict__ deg) {
    const long long gid = (long long)blockIdx.x * 256 + threadIdx.x;
    const int e    = (int)(gid >> 5);
    const int lane = (int)(gid & 31);
    if (e >= N_EDGES) return;
    const int src = ei[e];
    const int dst = ei[N_EDGES + e];

    const float4 v = *(const float4*)(h + (size_t)src * HID + lane * 4);
    float* dp = agg + (size_t)dst * HID + lane * 4;
    gatomic_add(dp + 0, v.x);
    gatomic_add(dp + 1, v.y);
    gatomic_add(dp + 2, v.z);
    gatomic_add(dp + 3, v.w);
    if (lane == 0) gatomic_add(deg + dst, 1.0f);
}

// ---------------------------------------------------------------------------
// Kernel 3: h2 = (agg/deg) @ lin_l^T + lin_l_b + h @ lin_r^T     (fused K=256)
//           then LayerNorm + ReLU per row, block-local pool sum in LDS, one
//           global atomic per feature per block. Post-LN tensor never stored.
// ---------------------------------------------------------------------------
__global__ __launch_bounds__(256) void k_conv(const float* __restrict__ agg,
                                              const float* __restrict__ deg,
                                              const float* __restrict__ h,
                                              const float* __restrict__ wl,
                                              const float* __restrict__ bl,
                                              const float* __restrict__ wrr,
                                              const float* __restrict__ lng,
                                              const float* __restrict__ lnb,
                                              float* __restrict__ pool) {
    __shared__ __align__(16) float smem[16][132];
    __shared__ float psum[HID];

    const int lane  = threadIdx.x & 31;
    const int wave  = threadIdx.x >> 5;
    const int row0  = blockIdx.x * 16;
    const int j0    = wave * 16;
    const int m     = lane & 15;
    const int koff4 = (lane >> 4) * 4;
    const int row   = row0 + m;

    const float invd = 1.0f / fmaxf(deg[row], 1.0f);
    const float* ar0 = agg + (size_t)row * HID + koff4;
    const float* ar1 = h   + (size_t)row * HID + koff4;
    const float* br0 = wl  + (size_t)(j0 + m) * HID + koff4;
    const float* br1 = wrr + (size_t)(j0 + m) * HID + koff4;

    v8f c = {0.f, 0.f, 0.f, 0.f, 0.f, 0.f, 0.f, 0.f};
#pragma unroll 4
    for (int k = 0; k < HID; k += 8) {          // mean_nbr @ lin_l^T
        v4f a4 = *(const v4f*)(ar0 + k);
        a4 *= invd;
        v4f b4 = *(const v4f*)(br0 + k);
        c = wmma_f32(lo2(a4), lo2(b4), c);
        c = wmma_f32(hi2(a4), hi2(b4), c);
    }
#pragma unroll 4
    for (int k = 0; k < HID; k += 8) {          // + h @ lin_r^T
        v4f a4 = *(const v4f*)(ar1 + k);
        v4f b4 = *(const v4f*)(br1 + k);
        c = wmma_f32(lo2(a4), lo2(b4), c);
        c = wmma_f32(hi2(a4), hi2(b4), c);
    }

    const int   col = j0 + m;
    const float bv  = bl[col];
    const int   rb  = (lane >> 4) * 8;
#pragma unroll
    for (int v = 0; v < 8; ++v) smem[rb + v][col] = c[v] + bv;

    if (threadIdx.x < HID) psum[threadIdx.x] = 0.f;
    __syncthreads();

    // LayerNorm + ReLU + pool: wave w handles rows 2w, 2w+1 (16 rows / 8 waves)
    const float4 g4 = *(const float4*)(lng + lane * 4);
    const float4 b4 = *(const float4*)(lnb + lane * 4);
#pragma unroll
    for (int rr = 0; rr < 2; ++rr) {
        const int r = wave * 2 + rr;
        const float4 vv = *(const float4*)&smem[r][lane * 4];
        float s  = vv.x + vv.y + vv.z + vv.w;
        float sq = vv.x * vv.x + vv.y * vv.y + vv.z * vv.z + vv.w * vv.w;
#pragma unroll
        for (int off = 16; off > 0; off >>= 1) {
            s  += __shfl_xor(s, off, 32);
            sq += __shfl_xor(sq, off, 32);
        }
        const float mu   = s * (1.0f / HID);
        const float var  = sq * (1.0f / HID) - mu * mu;
        const float rstd = rsqrtf(var + LN_EPS);
        const float o0 = fmaxf((vv.x - mu) * rstd * g4.x + b4.x, 0.f);
        const float o1 = fmaxf((vv.y - mu) * rstd * g4.y + b4.y, 0.f);
        const float o2 = fmaxf((vv.z - mu) * rstd * g4.z + b4.z, 0.f);
        const float o3 = fmaxf((vv.w - mu) * rstd * g4.w + b4.w, 0.f);
        atomicAdd(&psum[lane * 4 + 0], o0);
        atomicAdd(&psum[lane * 4 + 1], o1);
        atomicAdd(&psum[lane * 4 + 2], o2);
        atomicAdd(&psum[lane * 4 + 3], o3);
    }
    __syncthreads();
    if (threadIdx.x < HID) gatomic_add(pool + threadIdx.x, psum[threadIdx.x]);
}

// ---------------------------------------------------------------------------
// Kernel 4: out = (pool / N) @ out_w^T + out_b    (1x128 @ 128x128 — tiny)
// ---------------------------------------------------------------------------
__global__ __launch_bounds__(128) void k_out(const float* __restrict__ pool,
                                             const float* __restrict__ ow,
                                             const float* __restrict__ ob,
                                             float* __restrict__ out) {
    const int j = threadIdx.x;
    float acc = 0.f;
    const float scale = 1.0f / (float)N_NODES;
#pragma unroll 8
    for (int f = 0; f < HID; ++f) acc += (pool[f] * scale) * ow[j * HID + f];
    out[j] = acc + ob[j];
}

extern "C" void kernel_launch(void* const* d_in, const int* in_sizes, int n_in,
                              void* d_out, int out_size, void* d_ws, size_t ws_size,
                              hipStream_t stream) {
    const float* x      = (const float*)d_in[0];
    const int*   edge   = (const int*)d_in[1];
    const float* proj_w = (const float*)d_in[2];
    const float* proj_b = (const float*)d_in[3];
    const float* lin_lw = (const float*)d_in[4];
    const float* lin_lb = (const float*)d_in[5];
    const float* lin_rw = (const float*)d_in[6];
    const float* ln_g   = (const float*)d_in[7];
    const float* ln_b   = (const float*)d_in[8];
    const float* out_w  = (const float*)d_in[9];
    const float* out_b  = (const float*)d_in[10];
    float* out = (float*)d_out;

    // Workspace layout
    const size_t hBytes = (size_t)N_NODES * HID * sizeof(float);   // 25.6 MB
    char* ws   = (char*)d_ws;
    float* h   = (float*)(ws);
    float* agg = (float*)(ws + hBytes);
    float* deg = (float*)(ws + 2 * hBytes);
    float* pl  = (float*)(ws + 2 * hBytes + (size_t)N_NODES * sizeof(float));

    hipMemsetAsync(agg, 0, hBytes, stream);
    hipMemsetAsync(deg, 0, (size_t)N_NODES * sizeof(float), stream);
    hipMemsetAsync(pl,  0, HID * sizeof(float), stream);

    k_proj<<<N_NODES / 16, 256, 0, stream>>>(x, proj_w, proj_b, h);
    k_scatter<<<(N_EDGES * 32) / 256, 256, 0, stream>>>(edge, h, agg, deg);
    k_conv<<<N_NODES / 16, 256, 0, stream>>>(agg, deg, h, lin_lw, lin_lb, lin_rw,
                                             ln_g, ln_b, pl);
    k_out<<<1, 128, 0, stream>>>(pl, out_w, out_b, out);
}